// HeadAttentionMulti_34187939676688
// MI455X (gfx1250) — compile-verified
//
#include <hip/hip_runtime.h>
#include <hip/hip_bf16.h>

// MHA forward for B=8,S=2048,H=4,D=64 (DM=256), fp32.
// Memory-bound on the mandatory 537MB attn output -> keep full fp32 precision
// and use V_WMMA_F32_16X16X4_F32 (exact fp32 matrix op) for all four GEMMs.
// LDS staging now uses gfx1250 GLOBAL_LOAD_ASYNC_TO_LDS_B128 (ASYNCcnt path),
// avoiding the global->VGPR->ds_store round trip.

#define Bq 8
#define Sq 2048
#define Hq 4
#define Dq 64
#define DMq 256

typedef float v2f __attribute__((ext_vector_type(2)));
typedef float v8f __attribute__((ext_vector_type(8)));

static __device__ __forceinline__ v8f wmma_f32(v2f a, v2f b, v8f c) {
  return __builtin_amdgcn_wmma_f32_16x16x4_f32(
      /*neg_a=*/false, a, /*neg_b=*/false, b,
      /*c_mod=*/(short)0, c, /*reuse_a=*/false, /*reuse_b=*/false);
}

// Async copy of 16 bytes global -> LDS (per lane). LDS operand is the low 32
// bits of the generic pointer (ISA: generic LDS addr[31:0] is the LDS offset).
static __device__ __forceinline__ void async_cp16(void* lds_dst, const void* gsrc) {
  const unsigned ldsa = (unsigned)(unsigned long long)lds_dst;
  asm volatile("global_load_async_to_lds_b128 %0, %1, off"
               :: "v"(ldsa), "v"(gsrc)
               : "memory");
}
static __device__ __forceinline__ void async_wait0() {
  asm volatile("s_wait_asynccnt 0x0" ::: "memory");
}

// ---------------------------------------------------------------------------
// Kernel 1: fused QKV projection.  grid=(16384/16, 6), block=256 (8 waves).
// blockIdx.y selects a 128-col slab of the concatenated [Wq|Wk|Wv] output.
// Output layout: Q/K/V each (B,H,S,D).
// ---------------------------------------------------------------------------
__global__ __launch_bounds__(256) void k_qkv(
    const float* __restrict__ x,
    const float* __restrict__ Wq, const float* __restrict__ bq,
    const float* __restrict__ Wk, const float* __restrict__ bk,
    const float* __restrict__ Wv, const float* __restrict__ bv,
    float* __restrict__ Q, float* __restrict__ K, float* __restrict__ V) {
  __shared__ float xs[16 * DMq];  // 16KB: 16 tokens x 256 features
  const int t0 = blockIdx.x * 16;
  const int mat = blockIdx.y >> 1;              // 0=Q 1=K 2=V
  const int colbase = (blockIdx.y & 1) * 128;   // within the 256-wide matrix

  // async stage of the x tile: 1024 x 16B, 4 per thread
  for (int i = threadIdx.x; i < 16 * 64; i += blockDim.x)
    async_cp16(xs + i * 4, x + (size_t)t0 * DMq + i * 4);
  async_wait0();
  __syncthreads();

  const int lane = threadIdx.x & 31;
  const int wave = threadIdx.x >> 5;
  const int g = lane >> 4;       // lane group (K sub-offset for A/B frags)
  const int ln = lane & 15;      // M index for A, N index for B/C
  const int col = colbase + wave * 16;

  const float* W = (mat == 0) ? Wq : (mat == 1) ? Wk : Wv;
  const float* bias = (mat == 0) ? bq : (mat == 1) ? bk : bv;
  float* Out = (mat == 0) ? Q : (mat == 1) ? K : V;

  v8f acc = {};
  for (int k0 = 0; k0 < DMq; k0 += 4) {
    const int ka = k0 + 2 * g;
    v2f a, b;
    a.x = xs[ln * DMq + ka];
    a.y = xs[ln * DMq + ka + 1];
    b.x = W[(size_t)ka * DMq + col + ln];
    b.y = W[(size_t)(ka + 1) * DMq + col + ln];
    acc = wmma_f32(a, b, acc);
  }
  const float bs = bias[col + ln];
#pragma unroll
  for (int j = 0; j < 8; ++j) {
    const int t = t0 + j + 8 * g;            // token (C row M = j + 8*g)
    const int bb = t >> 11, s = t & (Sq - 1);
    const int feat = col + ln;               // C col N = ln
    const int h = feat >> 6, d = feat & 63;
    Out[(((size_t)bb * Hq + h) * Sq + s) * Dq + d] = acc[j] + bs;
  }
}

// ---------------------------------------------------------------------------
// Kernel 2: logits = Q K^T / sqrt(D) + mask bias.  grid=(B*H, S/16), block=256.
// Each of 8 waves sweeps 16 key-tiles; raw logits written to attn buffer.
// ---------------------------------------------------------------------------
__global__ __launch_bounds__(256) void k_logits(
    const float* __restrict__ Q, const float* __restrict__ K,
    const float* __restrict__ mask, float* __restrict__ attn) {
  __shared__ float qs[16 * Dq];
  __shared__ float mqs[16];
  const int bh = blockIdx.x;
  const int q0 = blockIdx.y * 16;
  const int bb = bh >> 2;

  // async stage Q tile: 16 x 64 floats = 256 x 16B, 1 per thread
  for (int i = threadIdx.x; i < 16 * 16; i += blockDim.x)
    async_cp16(qs + i * 4, Q + ((size_t)bh * Sq + q0) * Dq + i * 4);
  if (threadIdx.x < 16) mqs[threadIdx.x] = mask[(size_t)bb * Sq + q0 + threadIdx.x];
  async_wait0();
  __syncthreads();

  const int lane = threadIdx.x & 31;
  const int wave = threadIdx.x >> 5;
  const int g = lane >> 4;
  const int ln = lane & 15;

  for (int i = 0; i < 16; ++i) {
    const int kcol = (wave * 16 + i) * 16;
    // pull the next key-tile rows toward cache while WMMAs issue
    __builtin_prefetch(K + ((size_t)bh * Sq + kcol + 16 + ln) * Dq, 0, 0);
    v8f acc = {};
    const float* krow = K + ((size_t)bh * Sq + kcol + ln) * Dq;
    for (int d0 = 0; d0 < Dq; d0 += 4) {
      const int ka = d0 + 2 * g;
      v2f a, b;
      a.x = qs[ln * Dq + ka];
      a.y = qs[ln * Dq + ka + 1];
      b.x = krow[ka];          // B[k][n] = K[kcol+n][k]  (K^T)
      b.y = krow[ka + 1];
      acc = wmma_f32(a, b, acc);
    }
    const float mk = mask[(size_t)bb * Sq + kcol + ln];
    float* dst = attn + ((size_t)bh * Sq + q0) * Sq + kcol;
#pragma unroll
    for (int j = 0; j < 8; ++j) {
      const int qm = j + 8 * g;
      float mm = mqs[qm] + mk;
      mm = (mm > 1.f) ? 1.f : mm;
      dst[(size_t)qm * Sq + ln] = acc[j] * 0.125f + mm * (-1e9f);
    }
  }
}

// ---------------------------------------------------------------------------
// Kernel 3: row softmax in place.  grid = B*H*S rows, block=256 (8 el/thread).
// ---------------------------------------------------------------------------
__global__ __launch_bounds__(256) void k_softmax(float* __restrict__ attn) {
  __shared__ float red[256];
  float* row = attn + (size_t)blockIdx.x * Sq;
  float4* rv = (float4*)row;
  float4 v0 = rv[threadIdx.x * 2];
  float4 v1 = rv[threadIdx.x * 2 + 1];

  float m = fmaxf(fmaxf(fmaxf(v0.x, v0.y), fmaxf(v0.z, v0.w)),
                  fmaxf(fmaxf(v1.x, v1.y), fmaxf(v1.z, v1.w)));
  red[threadIdx.x] = m;
  __syncthreads();
  for (int s = 128; s > 0; s >>= 1) {
    if (threadIdx.x < s) red[threadIdx.x] = fmaxf(red[threadIdx.x], red[threadIdx.x + s]);
    __syncthreads();
  }
  m = red[0];
  __syncthreads();

  v0.x = __expf(v0.x - m); v0.y = __expf(v0.y - m);
  v0.z = __expf(v0.z - m); v0.w = __expf(v0.w - m);
  v1.x = __expf(v1.x - m); v1.y = __expf(v1.y - m);
  v1.z = __expf(v1.z - m); v1.w = __expf(v1.w - m);
  red[threadIdx.x] = (v0.x + v0.y + v0.z + v0.w) + (v1.x + v1.y + v1.z + v1.w);
  __syncthreads();
  for (int s = 128; s > 0; s >>= 1) {
    if (threadIdx.x < s) red[threadIdx.x] += red[threadIdx.x + s];
    __syncthreads();
  }
  const float inv = 1.0f / red[0];

  v0.x *= inv; v0.y *= inv; v0.z *= inv; v0.w *= inv;
  v1.x *= inv; v1.y *= inv; v1.z *= inv; v1.w *= inv;
  rv[threadIdx.x * 2] = v0;
  rv[threadIdx.x * 2 + 1] = v1;
}

// ---------------------------------------------------------------------------
// Kernel 4: ctx = attn @ V.  grid=(B*H, S/16), block=128 (4 waves, one 16-col
// slice of D each).  attn tiles async-staged through LDS in 16x256 chunks.
// ctx written in (B,S,DM) layout ready for the output projection.
// ---------------------------------------------------------------------------
__global__ __launch_bounds__(128) void k_ctx(
    const float* __restrict__ attn, const float* __restrict__ V,
    float* __restrict__ ctx) {
  __shared__ float as[16 * 256];  // 16KB
  const int bh = blockIdx.x;
  const int q0 = blockIdx.y * 16;
  const int bb = bh >> 2, hh = bh & 3;
  const int lane = threadIdx.x & 31;
  const int wave = threadIdx.x >> 5;
  const int g = lane >> 4;
  const int ln = lane & 15;

  v8f acc = {};
  for (int kc = 0; kc < Sq; kc += 256) {
    __syncthreads();  // previous chunk fully consumed
    for (int i = threadIdx.x; i < 16 * 64; i += blockDim.x) {
      const int r = i >> 6, c = i & 63;
      async_cp16(as + r * 256 + c * 4,
                 attn + ((size_t)bh * Sq + q0 + r) * Sq + kc + c * 4);
    }
    async_wait0();
    __syncthreads();
    for (int k0 = 0; k0 < 256; k0 += 4) {
      const int ka = k0 + 2 * g;
      v2f a, b;
      a.x = as[ln * 256 + ka];
      a.y = as[ln * 256 + ka + 1];
      b.x = V[((size_t)bh * Sq + kc + ka) * Dq + wave * 16 + ln];
      b.y = V[((size_t)bh * Sq + kc + ka + 1) * Dq + wave * 16 + ln];
      acc = wmma_f32(a, b, acc);
    }
  }
#pragma unroll
  for (int j = 0; j < 8; ++j) {
    const int M = j + 8 * g;
    ctx[((size_t)bb * Sq + q0 + M) * DMq + hh * Dq + wave * 16 + ln] = acc[j];
  }
}

// ---------------------------------------------------------------------------
// Kernel 5: out = ctx @ Wo + bo.  grid=(16384/16, 2), block=256 (8 waves).
// ---------------------------------------------------------------------------
__global__ __launch_bounds__(256) void k_outproj(
    const float* __restrict__ ctx, const float* __restrict__ Wo,
    const float* __restrict__ bo, float* __restrict__ out) {
  __shared__ float cs[16 * DMq];
  const int t0 = blockIdx.x * 16;
  const int colbase = blockIdx.y * 128;

  for (int i = threadIdx.x; i < 16 * 64; i += blockDim.x)
    async_cp16(cs + i * 4, ctx + (size_t)t0 * DMq + i * 4);
  async_wait0();
  __syncthreads();

  const int lane = threadIdx.x & 31;
  const int wave = threadIdx.x >> 5;
  const int g = lane >> 4;
  const int ln = lane & 15;
  const int col = colbase + wave * 16;

  v8f acc = {};
  for (int k0 = 0; k0 < DMq; k0 += 4) {
    const int ka = k0 + 2 * g;
    v2f a, b;
    a.x = cs[ln * DMq + ka];
    a.y = cs[ln * DMq + ka + 1];
    b.x = Wo[(size_t)ka * DMq + col + ln];
    b.y = Wo[(size_t)(ka + 1) * DMq + col + ln];
    acc = wmma_f32(a, b, acc);
  }
  const float bs = bo[col + ln];
#pragma unroll
  for (int j = 0; j < 8; ++j) {
    const int t = t0 + j + 8 * g;
    out[(size_t)t * DMq + col + ln] = acc[j] + bs;
  }
}

// ---------------------------------------------------------------------------
extern "C" void kernel_launch(void* const* d_in, const int* in_sizes, int n_in,
                              void* d_out, int out_size, void* d_ws, size_t ws_size,
                              hipStream_t stream) {
  const float* x    = (const float*)d_in[0];
  const float* mask = (const float*)d_in[1];
  const float* Wq   = (const float*)d_in[2];
  const float* bq   = (const float*)d_in[3];
  const float* Wk   = (const float*)d_in[4];
  const float* bk   = (const float*)d_in[5];
  const float* Wv   = (const float*)d_in[6];
  const float* bv   = (const float*)d_in[7];
  const float* Wo   = (const float*)d_in[8];
  const float* bo   = (const float*)d_in[9];

  float* out  = (float*)d_out;                          // (B,S,DM)
  float* attn = out + (size_t)Bq * Sq * DMq;            // (B,H,S,S)

  const size_t qkv_elems = (size_t)Bq * Hq * Sq * Dq;   // 4,194,304
  float* Q = (float*)d_ws;
  float* K = Q + qkv_elems;
  float* V = K + qkv_elems;
  float* ctx = Q;  // Q is dead after k_logits; reuse its slot (ws = 48MB)

  k_qkv<<<dim3(Bq * Sq / 16, 6), 256, 0, stream>>>(x, Wq, bq, Wk, bk, Wv, bv, Q, K, V);
  k_logits<<<dim3(Bq * Hq, Sq / 16), 256, 0, stream>>>(Q, K, mask, attn);
  k_softmax<<<dim3(Bq * Hq * Sq), 256, 0, stream>>>(attn);
  k_ctx<<<dim3(Bq * Hq, Sq / 16), 128, 0, stream>>>(attn, V, ctx);
  k_outproj<<<dim3(Bq * Sq / 16, 2), 256, 0, stream>>>(ctx, Wo, bo, out);
}